// Embedder_19378892439867
// MI455X (gfx1250) — compile-verified
//
#include <hip/hip_runtime.h>
#include <hip/hip_bf16.h>

typedef __attribute__((ext_vector_type(2))) float v2f;
typedef __attribute__((ext_vector_type(8))) float v8f;

#define D_MODEL   256
#define LT        60000          // common upsampled length == max_len
#define NTAPS     5
#define L250      15000
#define L500      30000
#define L1000     60000
#define DTILES    (D_MODEL / 16)                 // 16
#define CHUNK     256                            // t-values per wave (16 tiles)
#define NCHUNKS   ((LT + CHUNK - 1) / CHUNK)     // 235
#define WAVES_PER_BLOCK 8
#define XBLOCKS   ((NCHUNKS + WAVES_PER_BLOCK - 1) / WAVES_PER_BLOCK)  // 30
#define X_SIZE    ((size_t)3 * D_MODEL * LT)     // 46,080,000

// tap k of the K=5 conv lands on an input sample of stream with stride S iff (k-2)%S==0
__host__ __device__ constexpr bool tap_valid(int S, int k) {
    return (k < NTAPS) && (((k - 2) % S) == 0);
}
__host__ __device__ constexpr int tap_off(int S, int k) {
    return (k - 2) / S;   // truncation is exact for all valid taps here
}

template <int S>
__device__ inline void wave_conv(const float* __restrict__ x, int len,
                                 const float* __restrict__ W,
                                 const float* __restrict__ b,
                                 float* __restrict__ out,   // base of X[i] ([256][60000])
                                 int dtile, int t0, int lane)
{
    const int M  = lane & 15;     // A row within tile; B/C/D column (t) index
    const int hi = lane >> 4;     // lane half selects K-half (A/B) or M+8 (C/D)

    // ---- A operands: padded weight rows (loop invariant) ----
    float wv[8];
#pragma unroll
    for (int k = 0; k < 8; ++k)
        wv[k] = tap_valid(S, k) ? W[(dtile + M) * NTAPS + k] : 0.0f;
    v2f A1; A1.x = hi ? wv[2] : wv[0]; A1.y = hi ? wv[3] : wv[1];
    v2f A2; A2.x = hi ? wv[6] : wv[4]; A2.y = hi ? wv[7] : wv[5];

    // ---- C operand: bias broadcast, C[v] holds M = v + 8*hi ----
    v8f C;
#pragma unroll
    for (int v = 0; v < 8; ++v)
        C[v] = b[dtile + v + (hi ? 8 : 0)];

#pragma unroll 4
    for (int tt = 0; tt < CHUNK / 16; ++tt) {
        const int ttile = t0 + tt * 16;      // uniform per wave
        if (ttile >= LT) break;
        const int tn = ttile + M;            // this lane's t (B column / D column)

        // ---- B operands: input window, clamp+mask (no EXEC divergence) ----
        float sv[8];
#pragma unroll
        for (int k = 0; k < 8; ++k) {
            if (tap_valid(S, k)) {
                const int u  = tn + tap_off(S, k);
                const int uc = u < 0 ? 0 : (u >= len ? len - 1 : u);
                sv[k] = x[uc] * ((u >= 0 && u < len) ? 1.0f : 0.0f);
            } else {
                sv[k] = 0.0f;
            }
        }
        v2f B1; B1.x = hi ? sv[2] : sv[0]; B1.y = hi ? sv[3] : sv[1];
        v2f B2; B2.x = hi ? sv[6] : sv[4]; B2.y = hi ? sv[7] : sv[5];

        // D = W[:,4:8] * X1 + (W[:,0:4] * X0 + bias)
        v8f Dm = __builtin_amdgcn_wmma_f32_16x16x4_f32(
            false, A1, false, B1, (short)0, C, false, false);
        Dm = __builtin_amdgcn_wmma_f32_16x16x4_f32(
            false, A2, false, B2, (short)0, Dm, false, false);

        // ---- store (zero outside original stream length), streaming writes ----
        const float keep = (tn < len) ? 1.0f : 0.0f;
#pragma unroll
        for (int v = 0; v < 8; ++v) {
            const int m = dtile + v + (hi ? 8 : 0);
            __builtin_nontemporal_store(Dm[v] * keep, &out[(size_t)m * LT + tn]);
        }
    }
}

// grid = (XBLOCKS, DTILES, 3); block = 256 (8 waves, one 256-wide t-chunk each)
__global__ __launch_bounds__(256) void
embed_conv_wmma(const float* __restrict__ x250, const float* __restrict__ x500,
                const float* __restrict__ x1000, const float* __restrict__ W,
                const float* __restrict__ b, float* __restrict__ out)
{
    const int lane  = threadIdx.x & 31;
    const int chunk = blockIdx.x * WAVES_PER_BLOCK + (threadIdx.x >> 5);
    if (chunk >= NCHUNKS) return;             // uniform per wave

    const int dtile = blockIdx.y * 16;
    const int t0    = chunk * CHUNK;

    if (blockIdx.z == 0)
        wave_conv<4>(x250, L250, W, b, out, dtile, t0, lane);
    else if (blockIdx.z == 1)
        wave_conv<2>(x500, L500, W, b, out + (size_t)D_MODEL * LT, dtile, t0, lane);
    else
        wave_conv<1>(x1000, L1000, W, b, out + (size_t)2 * D_MODEL * LT, dtile, t0, lane);
}

// S = [rows; cols]: first 15000 entries from stream0, next 30000 from stream1,
// remaining 15000 from stream2 (total 105000 truncated to max_len = 60000).
__global__ __launch_bounds__(256) void
embed_s_kernel(float* __restrict__ out)
{
    const int t = blockIdx.x * blockDim.x + threadIdx.x;
    if (t >= LT) return;
    float row, col;
    if (t < L250)                { row = 0.0f; col = (float)(4 * t); }
    else if (t < L250 + L500)    { row = 1.0f; col = (float)(2 * (t - L250)); }
    else                         { row = 2.0f; col = (float)(t - (L250 + L500)); }
    __builtin_nontemporal_store(row, &out[X_SIZE + t]);
    __builtin_nontemporal_store(col, &out[X_SIZE + LT + t]);
}

extern "C" void kernel_launch(void* const* d_in, const int* in_sizes, int n_in,
                              void* d_out, int out_size, void* d_ws, size_t ws_size,
                              hipStream_t stream)
{
    (void)in_sizes; (void)n_in; (void)out_size; (void)d_ws; (void)ws_size;
    const float* x250  = (const float*)d_in[0];
    const float* x500  = (const float*)d_in[1];
    const float* x1000 = (const float*)d_in[2];
    const float* W     = (const float*)d_in[3];   // [256][1][5] row-major
    const float* b     = (const float*)d_in[4];   // [256]
    float* out = (float*)d_out;

    dim3 grid(XBLOCKS, DTILES, 3);
    embed_conv_wmma<<<grid, 256, 0, stream>>>(x250, x500, x1000, W, b, out);

    embed_s_kernel<<<(LT + 255) / 256, 256, 0, stream>>>(out);
}